// Model_23862838297147
// MI455X (gfx1250) — compile-verified
//
#include <hip/hip_runtime.h>
#include <math.h>

// Problem constants (shapes are fixed by the reference; N/E derived from in_sizes)
#define F_IN 128
#define HD   128     // H*D
#define NHEAD 4
#define DDIM  32
#define NEG_SLOPE 0.2f

typedef __attribute__((ext_vector_type(2))) float v2f;
typedef __attribute__((ext_vector_type(8))) float v8f;

// ---------------------------------------------------------------------------
// K0: initialize accumulators each call (harness poisons once, never re-poisons)
//   out   <- 0                 (N*128 floats)
//   denom <- 0                 (N*H floats)
//   emax  <- -inf (0xFF800000) (N*H cells, int-ordered float encoding)
// ---------------------------------------------------------------------------
__global__ void gat_init(float* __restrict__ out, float* __restrict__ denom,
                         unsigned* __restrict__ emax, int nOut, int nNH) {
  int i = blockIdx.x * blockDim.x + threadIdx.x;
  int stride = gridDim.x * blockDim.x;
  for (int j = i; j < nOut; j += stride) out[j] = 0.0f;
  for (int j = i; j < nNH; j += stride) {
    denom[j] = 0.0f;
    emax[j] = 0xFF800000u;  // -inf bit pattern
  }
}

// ---------------------------------------------------------------------------
// K1: feat_src = feat @ fc_w via V_WMMA_F32_16X16X4_F32
//   block = 256 threads = 8 waves; block owns 16 rows (M), wave w owns N-tile 16w
//   A tile (16x128 f32) staged in LDS, padded row stride 132 to avoid bank
//   conflicts (addr = r*132 + k -> banks (k + 4r) % 64, distinct for r=0..15).
//   Fragment layouts per CDNA5 ISA 7.12.2:
//     A 16x4 f32 : v0 = K0 (lanes 0-15) / K2 (lanes 16-31), v1 = K1 / K3
//     B  4x16 f32: mirrored (v0 = row K0 / K2 across N lanes)
//     C/D 16x16  : VGPR i, lanes 0-15 = row i, lanes 16-31 = row i+8
// ---------------------------------------------------------------------------
__global__ __launch_bounds__(256) void gat_gemm(const float* __restrict__ feat,
                                                const float* __restrict__ fcw,
                                                float* __restrict__ feat_src,
                                                int nNodes) {
  __shared__ float As[16 * 132];

  const int m0 = blockIdx.x * 16;
  const int tid = threadIdx.x;

  // Cooperative A-tile load: 16 rows x 128 cols = 512 float4, 2 per thread.
  for (int j = tid; j < 512; j += 256) {
    int row = j >> 5;          // 0..15
    int c4  = j & 31;          // 0..31 (float4 column)
    int gr  = m0 + row;
    if (gr >= nNodes) gr = nNodes - 1;  // safe clamp (N is a multiple of 16)
    float4 v = *reinterpret_cast<const float4*>(feat + (size_t)gr * F_IN + c4 * 4);
    *reinterpret_cast<float4*>(&As[row * 132 + c4 * 4]) = v;  // 528B row stride: 16B aligned
  }
  __syncthreads();

  const int wave = tid >> 5;
  const int lane = tid & 31;
  const int n0 = wave * 16;      // this wave's N-tile
  const int lm = lane & 15;      // row (A) / col (B) within tile
  const int hi = lane >> 4;      // K-half selector

  v8f acc = {0.f, 0.f, 0.f, 0.f, 0.f, 0.f, 0.f, 0.f};

#pragma unroll
  for (int k = 0; k < F_IN; k += 4) {
    const int kk = k + 2 * hi;
    v2f a;  // A fragment from LDS (ds_load_b64: 8B aligned since 132 even, kk even)
    a.x = As[lm * 132 + kk];
    a.y = As[lm * 132 + kk + 1];
    v2f b;  // B fragment straight from global (fc_w is 64KB, L2-resident)
    b.x = fcw[(size_t)kk * HD + n0 + lm];
    b.y = fcw[(size_t)(kk + 1) * HD + n0 + lm];
    acc = __builtin_amdgcn_wmma_f32_16x16x4_f32(
        /*neg_a=*/false, a, /*neg_b=*/false, b,
        /*c_mod=*/(short)0, acc, /*reuse_a=*/false, /*reuse_b=*/false);
  }

  // Write C tile: VGPR i -> row (m0 + hi*8 + i), col (n0 + lm)
#pragma unroll
  for (int i = 0; i < 8; ++i) {
    int gr = m0 + hi * 8 + i;
    if (gr < nNodes) feat_src[(size_t)gr * HD + n0 + lm] = acc[i];
  }
}

// ---------------------------------------------------------------------------
// K2: per-(node,head) attention logits: el = <feat_src[n,h,:], attn_l[h,:]>
//     thread i = node*H + h reads 32 contiguous floats (feat_src offset i*32)
// ---------------------------------------------------------------------------
__global__ void gat_elr(const float* __restrict__ feat_src,
                        const float* __restrict__ attn_l,
                        const float* __restrict__ attn_r,
                        float* __restrict__ el, float* __restrict__ er, int nNH) {
  int i = blockIdx.x * blockDim.x + threadIdx.x;
  if (i >= nNH) return;
  int h = i & (NHEAD - 1);
  const float* fp = feat_src + (size_t)i * DDIM;  // node*128 + h*32 == i*32
  const float* al = attn_l + h * DDIM;
  const float* ar = attn_r + h * DDIM;
  float sl = 0.f, sr = 0.f;
#pragma unroll
  for (int d = 0; d < DDIM; ++d) {
    float f = fp[d];
    sl += f * al[d];
    sr += f * ar[d];
  }
  el[i] = sl;
  er[i] = sr;
}

// int-ordered float atomic max (cell pre-initialized to -inf)
__device__ inline void atomicMaxF(unsigned* addr, float val) {
  if (val >= 0.f)
    atomicMax(reinterpret_cast<int*>(addr), __float_as_int(val));
  else
    atomicMin(addr, __float_as_uint(val));
}

// ---------------------------------------------------------------------------
// K3: per-edge LeakyReLU logits + segment max over dst
// ---------------------------------------------------------------------------
__global__ void gat_edge_logits(const int* __restrict__ src, const int* __restrict__ dst,
                                const float* __restrict__ el, const float* __restrict__ er,
                                float* __restrict__ ebuf, unsigned* __restrict__ emax,
                                int nEdges) {
  int e = blockIdx.x * blockDim.x + threadIdx.x;
  if (e >= nEdges) return;
  int s = src[e], d = dst[e];
  float4 l = *reinterpret_cast<const float4*>(el + (size_t)s * NHEAD);
  float4 r = *reinterpret_cast<const float4*>(er + (size_t)d * NHEAD);
  float v[NHEAD] = {l.x + r.x, l.y + r.y, l.z + r.z, l.w + r.w};
  float4 o;
#pragma unroll
  for (int h = 0; h < NHEAD; ++h) {
    float x = v[h];
    x = (x > 0.f) ? x : NEG_SLOPE * x;
    v[h] = x;
    atomicMaxF(&emax[(size_t)d * NHEAD + h], x);
  }
  o.x = v[0]; o.y = v[1]; o.z = v[2]; o.w = v[3];
  *reinterpret_cast<float4*>(ebuf + (size_t)e * NHEAD) = o;
}

// ---------------------------------------------------------------------------
// K4: ex = exp(e - emax[dst]) (in place) + segment sum into denom
// ---------------------------------------------------------------------------
__global__ void gat_exp(const int* __restrict__ dst, const unsigned* __restrict__ emax,
                        float* __restrict__ ebuf, float* __restrict__ denom, int nEdges) {
  int e = blockIdx.x * blockDim.x + threadIdx.x;
  if (e >= nEdges) return;
  int d = dst[e];
  float4 v = *reinterpret_cast<const float4*>(ebuf + (size_t)e * NHEAD);
  float ex[NHEAD];
  ex[0] = __expf(v.x - __uint_as_float(emax[(size_t)d * NHEAD + 0]));
  ex[1] = __expf(v.y - __uint_as_float(emax[(size_t)d * NHEAD + 1]));
  ex[2] = __expf(v.z - __uint_as_float(emax[(size_t)d * NHEAD + 2]));
  ex[3] = __expf(v.w - __uint_as_float(emax[(size_t)d * NHEAD + 3]));
  float4 o = {ex[0], ex[1], ex[2], ex[3]};
  *reinterpret_cast<float4*>(ebuf + (size_t)e * NHEAD) = o;
#pragma unroll
  for (int h = 0; h < NHEAD; ++h)
    atomicAdd(&denom[(size_t)d * NHEAD + h], ex[h]);
}

// ---------------------------------------------------------------------------
// K5: a = ex/denom[dst]; out[dst] += feat_src[src] * a
//   One wave32 per edge; lane owns one float4 of the 128-wide message
//   (lane -> head = lane>>3). Gather + atomic scatter are contiguous 512B/wave.
// ---------------------------------------------------------------------------
__global__ __launch_bounds__(256) void gat_scatter(const int* __restrict__ src,
                                                   const int* __restrict__ dst,
                                                   const float* __restrict__ ebuf,
                                                   const float* __restrict__ denom,
                                                   const float* __restrict__ feat_src,
                                                   float* __restrict__ out, int nEdges) {
  int gtid = blockIdx.x * blockDim.x + threadIdx.x;
  int e = gtid >> 5;          // one wave per edge
  int lane = threadIdx.x & 31;
  if (e >= nEdges) return;
  int s = src[e], d = dst[e];
  int h = lane >> 3;          // 8 lanes (32 floats) per head
  float a = ebuf[(size_t)e * NHEAD + h] / denom[(size_t)d * NHEAD + h];
  float4 f = *reinterpret_cast<const float4*>(feat_src + (size_t)s * HD + lane * 4);
  float* op = out + (size_t)d * HD + lane * 4;
  atomicAdd(op + 0, f.x * a);
  atomicAdd(op + 1, f.y * a);
  atomicAdd(op + 2, f.z * a);
  atomicAdd(op + 3, f.w * a);
}

// ---------------------------------------------------------------------------
// Launch
// ---------------------------------------------------------------------------
extern "C" void kernel_launch(void* const* d_in, const int* in_sizes, int n_in,
                              void* d_out, int out_size, void* d_ws, size_t ws_size,
                              hipStream_t stream) {
  const float* feat   = (const float*)d_in[0];
  const float* fc_w   = (const float*)d_in[1];
  const float* attn_l = (const float*)d_in[2];
  const float* attn_r = (const float*)d_in[3];
  const int*   src    = (const int*)d_in[4];
  const int*   dst    = (const int*)d_in[5];
  float* out = (float*)d_out;

  const int N = in_sizes[0] / F_IN;   // 100000
  const int E = in_sizes[4];          // 1600000
  const int NH = N * NHEAD;

  // Workspace carve-up (~83 MB total):
  float*    feat_src = (float*)d_ws;                    // N*128
  float*    el       = feat_src + (size_t)N * HD;       // N*H
  float*    er       = el + NH;                         // N*H
  float*    denom    = er + NH;                         // N*H
  unsigned* emax     = (unsigned*)(denom + NH);         // N*H
  float*    ebuf     = (float*)(emax + NH);             // E*H (logits, then ex in place)

  // K0: init out/denom/emax
  {
    int total = N * HD;
    int blocks = 2048;
    gat_init<<<blocks, 256, 0, stream>>>(out, denom, emax, total, NH);
  }
  // K1: GEMM (16 rows per block)
  {
    int blocks = (N + 15) / 16;
    gat_gemm<<<blocks, 256, 0, stream>>>(feat, fc_w, feat_src, N);
  }
  // K2: el/er
  {
    int blocks = (NH + 255) / 256;
    gat_elr<<<blocks, 256, 0, stream>>>(feat_src, attn_l, attn_r, el, er, NH);
  }
  // K3: edge logits + segment max
  {
    int blocks = (E + 255) / 256;
    gat_edge_logits<<<blocks, 256, 0, stream>>>(src, dst, el, er, ebuf, emax, E);
  }
  // K4: exp + segment sum
  {
    int blocks = (E + 255) / 256;
    gat_exp<<<blocks, 256, 0, stream>>>(dst, emax, ebuf, denom, E);
  }
  // K5: normalize + weighted scatter-sum (1 wave per edge, 8 waves per block)
  {
    int blocks = (E + 7) / 8;
    gat_scatter<<<blocks, 256, 0, stream>>>(src, dst, ebuf, denom, feat_src, out, E);
  }
}